// SAGE_18820546691594
// MI455X (gfx1250) — compile-verified
//
#include <hip/hip_runtime.h>
#include <math.h>

typedef __attribute__((ext_vector_type(2))) float v2f;
typedef __attribute__((ext_vector_type(8))) float v8f;

// -------- constants matching the reference --------
#define N0_  262144
#define INF_ 128
#define H_   32
#define N1_  131072
#define N2_  65536

__device__ __forceinline__ v8f wmma_f32_k4(v2f a, v2f b, v8f c) {
  // V_WMMA_F32_16X16X4_F32 : D = A(16x4) * B(4x16) + C(16x16), all f32
  return __builtin_amdgcn_wmma_f32_16x16x4_f32(
      /*neg_a=*/false, a, /*neg_b=*/false, b,
      /*c_mod=*/(short)0, c, /*reuse_a=*/false, /*reuse_b=*/false);
}

// ---------------- edge scatter-add (mean aggregation numerator + counts) ----
// One thread handles one float4 group of one edge. F4 = row_width/4.
template <int F4, int LOG2F4>
__global__ __launch_bounds__(256) void sage_scatter_kernel(
    const float* __restrict__ xsrc, const int* __restrict__ src,
    const int* __restrict__ tgt, float* __restrict__ agg,
    float* __restrict__ cnt, int nedges) {
  long long tid = (long long)blockIdx.x * blockDim.x + threadIdx.x;
  long long total = (long long)nedges << LOG2F4;
  if (tid >= total) return;
  int e = (int)(tid >> LOG2F4);
  int g = (int)(tid & (F4 - 1));
  int s = src[e];
  int t = tgt[e];
  const float4* x4 = (const float4*)xsrc;
  float4 v = x4[(size_t)s * F4 + g];
  float* dst = agg + (size_t)t * (F4 * 4) + g * 4;
  unsafeAtomicAdd(dst + 0, v.x);
  unsafeAtomicAdd(dst + 1, v.y);
  unsafeAtomicAdd(dst + 2, v.z);
  unsafeAtomicAdd(dst + 3, v.w);
  if (g == 0) unsafeAtomicAdd(cnt + t, 1.0f);
}

// ---------------- per-node transform via WMMA ------------------------------
// out[i,:] = l2norm( (agg[i,:]/max(cnt,1)) @ wl^T + bl + xin[i,:] @ wr^T )
// One wave32 computes a 16-row tile; H=32 output cols = two 16x16 C tiles.
template <int K, bool RELU>
__global__ __launch_bounds__(256) void sage_compute_kernel(
    const float* __restrict__ agg, const float* __restrict__ cnt,
    const float* __restrict__ xin, const float* __restrict__ wl,
    const float* __restrict__ bl, const float* __restrict__ wr,
    float* __restrict__ out, int nrows) {
  const int lane = threadIdx.x & 31;
  const int wid  = blockIdx.x * (blockDim.x >> 5) + (threadIdx.x >> 5);
  const int base = wid * 16;
  if (base >= nrows) return;            // wave-uniform: EXEC stays all-ones
  const int l16  = lane & 15;
  const int half = lane >> 4;           // 0: K even pair / rows r ; 1: K odd pair / rows r+8
  const int rowA = base + l16;

  const float inv = 1.0f / fmaxf(cnt[rowA], 1.0f);

  v8f c0 = {};  // cols 0..15
  v8f c1 = {};  // cols 16..31

#pragma unroll
  for (int phase = 0; phase < 2; ++phase) {
    const float* Arow = (phase == 0 ? agg : xin) + (size_t)rowA * K;
    const float* W    = (phase == 0 ? wl : wr);
    const float  s    = (phase == 0 ? inv : 1.0f);
    const float2* A2  = (const float2*)Arow;                  // per-lane row
    const float2* B0  = (const float2*)(W + (size_t)l16 * K);        // col l16
    const float2* B1  = (const float2*)(W + (size_t)(l16 + 16) * K); // col l16+16
#pragma unroll
    for (int kc = 0; kc < K; kc += 4) {
      const int fi = (kc >> 1) + half;  // lanes<16: K=kc,kc+1 ; lanes>=16: kc+2,kc+3
      float2 fa = A2[fi];
      float2 f0 = B0[fi];
      float2 f1 = B1[fi];
      v2f a;  a.x  = fa.x * s; a.y  = fa.y * s;
      v2f b0; b0.x = f0.x;     b0.y = f0.y;
      v2f b1; b1.x = f1.x;     b1.y = f1.y;
      c0 = wmma_f32_k4(a, b0, c0);
      c1 = wmma_f32_k4(a, b1, c1);
    }
  }

  // bias (per output column)
  const float bias0 = bl[l16];
  const float bias1 = bl[l16 + 16];
#pragma unroll
  for (int r = 0; r < 8; ++r) { c0[r] += bias0; c1[r] += bias1; }

  // per-row L2 normalize: row (base+r+8*half) lives in lanes of this half
#pragma unroll
  for (int r = 0; r < 8; ++r) {
    float p = c0[r] * c0[r] + c1[r] * c1[r];
    p += __shfl_xor(p, 1, 32);
    p += __shfl_xor(p, 2, 32);
    p += __shfl_xor(p, 4, 32);
    p += __shfl_xor(p, 8, 32);          // stays within each 16-lane half
    float scale = 1.0f / fmaxf(sqrtf(p), 1e-12f);
    float o0 = c0[r] * scale;
    float o1 = c1[r] * scale;
    if (RELU) { o0 = fmaxf(o0, 0.0f); o1 = fmaxf(o1, 0.0f); }
    const size_t row = (size_t)(base + r + half * 8);
    out[row * 32 + l16]      = o0;
    out[row * 32 + 16 + l16] = o1;
  }
}

extern "C" void kernel_launch(void* const* d_in, const int* in_sizes, int n_in,
                              void* d_out, int out_size, void* d_ws,
                              size_t ws_size, hipStream_t stream) {
  const float* x   = (const float*)d_in[0];
  const int* src0  = (const int*)d_in[1];
  const int* tgt0  = (const int*)d_in[2];
  const int* src1  = (const int*)d_in[3];
  const int* tgt1  = (const int*)d_in[4];
  const float* wl0 = (const float*)d_in[5];
  const float* bl0 = (const float*)d_in[6];
  const float* wr0 = (const float*)d_in[7];
  const float* wl1 = (const float*)d_in[8];
  const float* bl1 = (const float*)d_in[9];
  const float* wr1 = (const float*)d_in[10];
  const int E0 = in_sizes[1];
  const int E1 = in_sizes[3];

  // workspace layout (floats): agg0 | cnt0 | agg1 | cnt1 | h
  float* ws   = (float*)d_ws;
  float* agg0 = ws;
  float* cnt0 = agg0 + (size_t)N1_ * INF_;
  float* agg1 = cnt0 + (size_t)N1_;
  float* cnt1 = agg1 + (size_t)N2_ * H_;
  float* h    = cnt1 + (size_t)N2_;
  const size_t zero_bytes =
      ((size_t)N1_ * INF_ + N1_ + (size_t)N2_ * H_ + N2_) * sizeof(float);
  hipMemsetAsync(d_ws, 0, zero_bytes, stream);

  // ---- layer 0: scatter-add x[src0] into agg0[tgt0], count edges ----
  {
    long long total = (long long)E0 * 32;  // 128 floats -> 32 float4 groups
    int blocks = (int)((total + 255) / 256);
    sage_scatter_kernel<32, 5><<<blocks, 256, 0, stream>>>(x, src0, tgt0, agg0,
                                                           cnt0, E0);
  }
  // ---- layer 0: h = relu(l2norm(mean@wl0^T + bl0 + x[:N1]@wr0^T)) ----
  {
    int blocks = (N1_ / 16) / 8;  // 8 waves per 256-thread block
    sage_compute_kernel<INF_, true><<<blocks, 256, 0, stream>>>(
        agg0, cnt0, x, wl0, bl0, wr0, h, N1_);
  }
  // ---- layer 1: scatter-add h[src1] into agg1[tgt1] ----
  {
    long long total = (long long)E1 * 8;  // 32 floats -> 8 float4 groups
    int blocks = (int)((total + 255) / 256);
    sage_scatter_kernel<8, 3><<<blocks, 256, 0, stream>>>(h, src1, tgt1, agg1,
                                                          cnt1, E1);
  }
  // ---- layer 1: out = l2norm(mean@wl1^T + bl1 + h[:N2]@wr1^T) ----
  {
    int blocks = (N2_ / 16) / 8;
    sage_compute_kernel<H_, false><<<blocks, 256, 0, stream>>>(
        agg1, cnt1, h, wl1, bl1, wr1, (float*)d_out, N2_);
  }
}